// CategoricalDiffusion_4380866642588
// MI455X (gfx1250) — compile-verified
//
#include <hip/hip_runtime.h>

typedef __attribute__((ext_vector_type(2))) float v2f;
typedef __attribute__((ext_vector_type(4))) float v4f;
typedef __attribute__((ext_vector_type(8))) float v8f;

#define WAVES 8
#define GROUPS_PER_WAVE 4
#define PX_PER_BLOCK (WAVES * GROUPS_PER_WAVE * 16)   // 512 pixels per block

// Deterministic uniform(0,1) hash (seed 42 folded in); stands in for JAX gumbel noise.
__device__ __forceinline__ float hash_u01(unsigned x) {
  x = x * 0x9E3779B9u + 42u;
  x ^= x >> 16; x *= 0x85EBCA6Bu;
  x ^= x >> 13; x *= 0xC2B2AE35u;
  x ^= x >> 16;
  return ((float)(x >> 8) + 0.5f) * (1.0f / 16777216.0f);
}

// Intra-wave LDS producer->consumer sync (per-wave tiles, no block barrier needed).
__device__ __forceinline__ void wave_lds_sync() {
  asm volatile("s_wait_dscnt 0x0" ::: "memory");
  __builtin_amdgcn_wave_barrier();
}

__global__ __launch_bounds__(256, 2)
void catdiff_qrev_kernel(const int*   __restrict__ x_t,
                         const float* __restrict__ pred,
                         const int*   __restrict__ t,
                         const float* __restrict__ Qs,
                         const float* __restrict__ Qbs,
                         float*       __restrict__ out_idx,
                         float*       __restrict__ out_anc,
                         int pix_per_batch, int blocks_per_batch)
{
  // Tables padded to 16 columns (cols 8..15 == 0) => branch-free gathers everywhere.
  __shared__ float sRightP[8 * 16];     // sRightP[x*16+c] = Qbs[tb-1, x, c] (c<8), else 0
  __shared__ float sLeftP [8 * 16];     // sLeftP [v*16+c] = Qs[tb, c, v]   (c<8), else 0
  __shared__ float sInvD  [64];         // sInvD[v*8+x] = 1 / sum_c left_v[c]*right[x,c]
  __shared__ float sTile[WAVES][256];   // 16x16 raw-s tile per wave (D-layout -> row-major)

  const int tid = threadIdx.x;
  const int b   = blockIdx.x / blocks_per_batch;
  const int blk = blockIdx.x % blocks_per_batch;
  const int tb  = t[b];

  // ---- per-block tables (Qs/Qbs rows are tiny and L2-resident) ----
  if (tid < 128) {
    const int row = tid >> 4, col = tid & 15;   // row = x (right) / v (left)
    sRightP[tid] = (col < 8) ? Qbs[(tb - 1) * 64 + row * 8 + col] : 0.f;
    sLeftP [tid] = (col < 8) ? Qs[tb * 64 + col * 8 + row]        : 0.f;
  }
  __syncthreads();
  if (tid < 64) {
    const int v = tid >> 3, x = tid & 7;
    float s = 0.f;
    #pragma unroll
    for (int c = 0; c < 8; ++c) s += sLeftP[v * 16 + c] * sRightP[x * 16 + c];
    sInvD[tid] = __builtin_amdgcn_rcpf(s);      // raw v_rcp_f32
  }
  __syncthreads();

  const int wave = tid >> 5;
  const int lane = tid & 31;
  const int h    = lane >> 4;   // lane half: owns contiguous channels 4h..4h+3
  const int p    = lane & 15;   // pixel index (A phase) / column index (B & D phases)

  // Channel permutation: WMMA chunk0 reduces channels {0,1,4,5}, chunk1 {2,3,6,7}
  // (reduction order is free as long as A and B use the same permutation).
  // A layout: VGPR0 -> K=2h+0-slot, VGPR1 -> K=2h+1-slot; this lane supplies
  // channels 4h (a0.x), 4h+1 (a0.y), 4h+2 (a1.x), 4h+3 (a1.y).
  // B rows must match: b0 = right rows (4h, 4h+1), b1 = rows (4h+2, 4h+3); col = lane&15.
  v2f b0, b1;
  b0.x = sRightP[(4 * h + 0) * 16 + p];
  b0.y = sRightP[(4 * h + 1) * 16 + p];
  b1.x = sRightP[(4 * h + 2) * 16 + p];
  b1.y = sRightP[(4 * h + 3) * 16 + p];

  const long long batch_base = (long long)b * pix_per_batch;

  for (int it = 0; it < GROUPS_PER_WAVE; ++it) {
    const int group   = it * WAVES + wave;
    const int pixbase = blk * PX_PER_BLOCK + group * 16;
    if (pixbase + 16 > pix_per_batch) break;           // wave-uniform guard (EXEC stays full)
    const long long gp = batch_base + pixbase;

    if (it + 1 < GROUPS_PER_WAVE) {                    // global_prefetch_b8 for next group
      __builtin_prefetch(pred + (gp + WAVES * 16 + p) * 8, 0, 3);
    }

    const int v = x_t[gp + p];                         // lane L holds v for pixel L&15

    // one b128 per lane: this lane's 4 contiguous channels of its pixel
    const v4f pv = ((const v4f*)(pred + (gp + p) * 8))[h];

    // 8-wide softmax split across lane pairs (L, L^16)
    float m = fmaxf(fmaxf(pv.x, pv.y), fmaxf(pv.z, pv.w));
    m = fmaxf(m, __shfl_xor(m, 16, 32));
    float e0 = __expf(pv.x - m), e1 = __expf(pv.y - m);
    float e2 = __expf(pv.z - m), e3 = __expf(pv.w - m);
    float ssum = e0 + e1 + e2 + e3;
    ssum += __shfl_xor(ssum, 16, 32);
    const float sinv = __builtin_amdgcn_rcpf(ssum);

    // w[x] = softmax[x] / denom_v[x]; invD for 4 contiguous channels = one aligned b128
    const v4f id4 = *(const v4f*)&sInvD[v * 8 + 4 * h];
    v2f a0, a1;
    a0.x = e0 * sinv * id4.x;
    a0.y = e1 * sinv * id4.y;
    a1.x = e2 * sinv * id4.z;
    a1.y = e3 * sinv * id4.w;

    // s[p][c] = sum_x w[p][x] * right[x][c]   (two K=4 chunks, f32 WMMA)
    v8f acc = {0.f, 0.f, 0.f, 0.f, 0.f, 0.f, 0.f, 0.f};
    acc = __builtin_amdgcn_wmma_f32_16x16x4_f32(false, a0, false, b0, (short)0, acc, false, false);
    acc = __builtin_amdgcn_wmma_f32_16x16x4_f32(false, a1, false, b1, (short)0, acc, false, false);

    // store raw s tile; D layout: VGPR r holds (M=r+8h, N=lane&15)
    #pragma unroll
    for (int r = 0; r < 8; ++r) {
      sTile[wave][(r + 8 * h) * 16 + p] = acc[r];
    }

    wave_lds_sync();   // tile complete

    // Read back own pixel row (4 channels) and apply left[v][c] scale here:
    // this lane's v IS v_p for its pixel p, so no cross-lane traffic needed.
    const v4f sv = ((const v4f*)&sTile[wave][0])[p * 4 + h];
    const v4f lv = ((const v4f*)&sLeftP[v * 16])[h];
    v4f av = sv * lv;                                  // anc[p][4h..4h+3]

    // coalesced ancestral store: each lane writes a float4 of its pixel row (512B/wave)
    ((v4f*)(out_anc + (gp + p) * 8))[h] = av;

    // Gumbel argmax without logs of anc:
    //   argmax_c( log anc_c + g_c ) == argmax_c( anc_c / E_c ),  E_c = -log u_c
    // and a constant positive factor is argmax-invariant, so raw v_log_f32 (log2) works.
    const unsigned base = (unsigned)((gp + p) * 8) + 4 * (unsigned)h;
    float anc4[4] = {av.x, av.y, av.z, av.w};
    float best = -1.f;
    int   bc   = 4 * h;
    #pragma unroll
    for (int j = 0; j < 4; ++j) {
      float u   = hash_u01(base + (unsigned)j);
      float E   = -__builtin_amdgcn_logf(u);           // -log2(u) > 0, no denorm path
      float val = anc4[j] * __builtin_amdgcn_rcpf(E);
      if (j == 0 || val > best) { best = val; bc = 4 * h + j; }
    }
    // combine the two half-lane candidates (lower channel wins ties, like jnp.argmax)
    const float ob = __shfl_xor(best, 16, 32);
    const int   oc = __shfl_xor(bc,   16, 32);
    if (ob > best || (ob == best && oc < bc)) bc = oc;
    if (h == 0) out_idx[gp + p] = (float)bc;           // 16 consecutive floats per wave
  }
}

extern "C" void kernel_launch(void* const* d_in, const int* in_sizes, int n_in,
                              void* d_out, int out_size, void* d_ws, size_t ws_size,
                              hipStream_t stream) {
  const int*   x_t  = (const int*)d_in[0];
  const float* pred = (const float*)d_in[1];
  const int*   t    = (const int*)d_in[2];
  const float* Qs   = (const float*)d_in[3];
  const float* Qbs  = (const float*)d_in[4];

  const int total_pix = in_sizes[0];       // N * n * n
  const int N         = in_sizes[2];       // batch
  const int pix_per_batch = total_pix / N;

  float* out_idx = (float*)d_out;                       // argmax indices (as float), N*n*n
  float* out_anc = (float*)d_out + (size_t)total_pix;   // ancestral_probs, N*n*n*8

  const int blocks_per_batch = (pix_per_batch + PX_PER_BLOCK - 1) / PX_PER_BLOCK;
  dim3 grid(N * blocks_per_batch), block(WAVES * 32);

  catdiff_qrev_kernel<<<grid, block, 0, stream>>>(x_t, pred, t, Qs, Qbs,
                                                  out_idx, out_anc,
                                                  pix_per_batch, blocks_per_batch);
}